// MultiHeadAttention_23381801959681
// MI455X (gfx1250) — compile-verified
//
#include <hip/hip_runtime.h>

// MHA for MI455X (gfx1250, wave32). fp32 reference -> exact-precision
// V_WMMA_F32_16X16X4_F32 matrix path. ~69 GFLOP total vs a 537MB mandatory
// attn writeback => near the 23.3 TB/s bandwidth floor. Double-buffered LDS
// staging (1 barrier/iter) with GLOBAL_LOAD_ASYNC_TO_LDS_B128 (ASYNCcnt) so
// global->LDS tile movement bypasses VGPRs and overlaps the WMMA pipe.

typedef float v2f __attribute__((ext_vector_type(2)));
typedef float v8f __attribute__((ext_vector_type(8)));
typedef int   v4i __attribute__((vector_size(16)));

#define S_LEN   2048
#define DM      1024
#define NHEADS  16
#define BATCH   2
#define ROWS    (BATCH * S_LEN)          // 4096

#if __has_builtin(__builtin_amdgcn_global_load_async_to_lds_b128)
#define ASYNC_LDS 1
#else
#define ASYNC_LDS 0
#endif

__device__ __forceinline__ void wait_asynccnt0()
{
#if __has_builtin(__builtin_amdgcn_s_wait_asynccnt)
    __builtin_amdgcn_s_wait_asynccnt(0);
#else
    asm volatile("s_wait_asynccnt 0x0" ::: "memory");
#endif
}

#if ASYNC_LDS
__device__ __forceinline__ void async_copy16(const float* g, float* l)
{
    // param types per hipcc diagnostic: AS1 ("__device__") v4i*, AS3 v4i*
    __builtin_amdgcn_global_load_async_to_lds_b128(
        (__attribute__((address_space(1))) v4i*)(g),
        (__attribute__((address_space(3))) v4i*)(l), 0, 0);
}
#endif

// ---------------------------------------------------------------------------
// Generic tiled WMMA fp32 GEMM:  C[z] = scale * A[z] x B[z] (+ bias[n])
// Row-major A (MxK). B is KxN row-major (TRANSB=0) or NxK row-major (TRANSB=1,
// i.e. C = A x B^T). Per-z two-level base offsets (batch b = z/H, head h = z%H).
// Block: 128 threads = 4 waves; block tile 64x64; wave tile 32x32 (4x v8f).
// Double-buffered LDS, next K-tile staged during current tile's WMMAs.
// ---------------------------------------------------------------------------
template <bool TRANSB>
__global__ __launch_bounds__(128) void gemm_wmma_f32(
    const float* __restrict__ A, int lda, long sAb, long sAh,
    const float* __restrict__ B, int ldb, long sBb, long sBh,
    float* __restrict__ C, int ldc, long sCb, long sCh,
    const float* __restrict__ bias, float scale,
    int M, int N, int K, int H)
{
    constexpr int BM = 64, BN = 64, KB = 16;
    constexpr int ASTR = 20;   // 16 cols + pad: 16B-aligned b128 slots, bank spread
    constexpr int BSTR = 68;   // KB x BN staging stride (NN path)
    constexpr int BS_SZ = TRANSB ? (BN * ASTR) : (KB * BSTR);

    __shared__ float As[2][BM * ASTR];
    __shared__ float Bs[2][BS_SZ];

    const int z  = blockIdx.z;
    const int bb = z / H;
    const int hh = z - bb * H;
    const float* Ab = A + (long)bb * sAb + (long)hh * sAh;
    const float* Bb = B + (long)bb * sBb + (long)hh * sBh;
    float*       Cb = C + (long)bb * sCb + (long)hh * sCh;

    const int tid  = threadIdx.x;
    const int lane = tid & 31;
    const int wave = tid >> 5;
    const int wm   = (wave >> 1) * 32;
    const int wn   = (wave & 1) * 32;
    const int m0   = blockIdx.y * BM;
    const int n0   = blockIdx.x * BN;

    // fragment lane mapping (ISA 7.12.2, 32-bit A 16x4 / B 4x16):
    //   lanes 0-15: K = {kk, kk+1}; lanes 16-31: K = {kk+2, kk+3}
    const int fr = lane & 15;            // fragment row (A) / col (B)
    const int kb = (lane >> 4) * 2;      // k sub-offset within 4-wide step

    // per-thread staging slots (2x float4 for A tile, 2x for B tile)
    const int ar0 = tid >> 2,           ac0 = (tid & 3) << 2;          // A slot 0
    const int ar1 = (tid + 128) >> 2,   ac1 = ac0;                     // A slot 1
    int br0, bc0, br1, bc1;
    if (TRANSB) { br0 = tid >> 2;  bc0 = (tid & 3) << 2;  br1 = (tid + 128) >> 2;  bc1 = bc0; }
    else        { br0 = tid >> 4;  bc0 = (tid & 15) << 2; br1 = (tid + 128) >> 4;  bc1 = bc0; }

    v8f acc[2][2] = {};

    // ---- tile staging -----------------------------------------------------
#if ASYNC_LDS
    auto stage = [&](int p, int k0) {
        async_copy16(Ab + (long)(m0 + ar0) * lda + k0 + ac0, &As[p][ar0 * ASTR + ac0]);
        async_copy16(Ab + (long)(m0 + ar1) * lda + k0 + ac1, &As[p][ar1 * ASTR + ac1]);
        if (TRANSB) {
            async_copy16(Bb + (long)(n0 + br0) * ldb + k0 + bc0, &Bs[p][br0 * ASTR + bc0]);
            async_copy16(Bb + (long)(n0 + br1) * ldb + k0 + bc1, &Bs[p][br1 * ASTR + bc1]);
        } else {
            async_copy16(Bb + (long)(k0 + br0) * ldb + n0 + bc0, &Bs[p][br0 * BSTR + bc0]);
            async_copy16(Bb + (long)(k0 + br1) * ldb + n0 + bc1, &Bs[p][br1 * BSTR + bc1]);
        }
    };
#else
    float4 ra0, ra1, rb0, rb1;
    auto ldtile = [&](int k0) {
        ra0 = *(const float4*)(Ab + (long)(m0 + ar0) * lda + k0 + ac0);
        ra1 = *(const float4*)(Ab + (long)(m0 + ar1) * lda + k0 + ac1);
        if (TRANSB) {
            rb0 = *(const float4*)(Bb + (long)(n0 + br0) * ldb + k0 + bc0);
            rb1 = *(const float4*)(Bb + (long)(n0 + br1) * ldb + k0 + bc1);
        } else {
            rb0 = *(const float4*)(Bb + (long)(k0 + br0) * ldb + n0 + bc0);
            rb1 = *(const float4*)(Bb + (long)(k0 + br1) * ldb + n0 + bc1);
        }
    };
    auto sttile = [&](int p) {
        *(float4*)(&As[p][ar0 * ASTR + ac0]) = ra0;
        *(float4*)(&As[p][ar1 * ASTR + ac1]) = ra1;
        if (TRANSB) {
            *(float4*)(&Bs[p][br0 * ASTR + bc0]) = rb0;
            *(float4*)(&Bs[p][br1 * ASTR + bc1]) = rb1;
        } else {
            *(float4*)(&Bs[p][br0 * BSTR + bc0]) = rb0;
            *(float4*)(&Bs[p][br1 * BSTR + bc1]) = rb1;
        }
    };
#endif

    auto compute = [&](int p) {
        #pragma unroll
        for (int kk = 0; kk < KB; kk += 4) {
            const v2f a0 = *(const v2f*)(&As[p][(wm +      fr) * ASTR + kk + kb]);
            const v2f a1 = *(const v2f*)(&As[p][(wm + 16 + fr) * ASTR + kk + kb]);
            v2f b0, b1;
            if (TRANSB) {
                b0 = *(const v2f*)(&Bs[p][(wn +      fr) * ASTR + kk + kb]);
                b1 = *(const v2f*)(&Bs[p][(wn + 16 + fr) * ASTR + kk + kb]);
            } else {
                b0.x = Bs[p][(kk + kb)     * BSTR + wn +      fr];
                b0.y = Bs[p][(kk + kb + 1) * BSTR + wn +      fr];
                b1.x = Bs[p][(kk + kb)     * BSTR + wn + 16 + fr];
                b1.y = Bs[p][(kk + kb + 1) * BSTR + wn + 16 + fr];
            }
            acc[0][0] = __builtin_amdgcn_wmma_f32_16x16x4_f32(false, a0, false, b0, (short)0, acc[0][0], false, false);
            acc[0][1] = __builtin_amdgcn_wmma_f32_16x16x4_f32(false, a0, false, b1, (short)0, acc[0][1], false, false);
            acc[1][0] = __builtin_amdgcn_wmma_f32_16x16x4_f32(false, a1, false, b0, (short)0, acc[1][0], false, false);
            acc[1][1] = __builtin_amdgcn_wmma_f32_16x16x4_f32(false, a1, false, b1, (short)0, acc[1][1], false, false);
        }
    };

    // ---- main loop: double-buffered, 1 barrier per K-tile ------------------
#if ASYNC_LDS
    stage(0, 0);
    int p = 0;
    for (int k0 = 0; k0 < K; k0 += KB, p ^= 1) {
        wait_asynccnt0();            // my async copies for tile p are in LDS
        __syncthreads();             // everyone's are
        if (k0 + KB < K) stage(p ^ 1, k0 + KB);   // overlaps with WMMAs below
        compute(p);
    }
#else
    ldtile(0);
    int p = 0;
    for (int k0 = 0; k0 < K; k0 += KB, p ^= 1) {
        sttile(p);
        __syncthreads();
        if (k0 + KB < K) ldtile(k0 + KB);          // in flight during WMMAs
        compute(p);
    }
#endif

    // ---- store: C/D layout -> VGPR r holds M = r + 8*(lane>>4), N = lane&15
    const int rhalf = (lane >> 4) * 8;
    #pragma unroll
    for (int i = 0; i < 2; ++i) {
        #pragma unroll
        for (int j = 0; j < 2; ++j) {
            const int col = n0 + wn + 16 * j + fr;
            const float bv = bias ? bias[col] : 0.0f;
            #pragma unroll
            for (int r = 0; r < 8; ++r) {
                const int row = m0 + wm + 16 * i + rhalf + r;
                Cb[(long)row * ldc + col] = acc[i][j][r] * scale + bv;
            }
        }
    }
}

// ---------------------------------------------------------------------------
// In-place row softmax over S_LEN=2048 columns. One 256-thread block per row.
// float4 vectorized (attn stream is the roofline-dominant 537MB of traffic),
// wave32 shuffle reductions + 8-entry LDS handoff.
// ---------------------------------------------------------------------------
__global__ __launch_bounds__(256) void softmax_rows(float* __restrict__ attn)
{
    __shared__ float red[8];
    const long row = blockIdx.x;
    float4* p = (float4*)(attn + row * (long)S_LEN);
    const int tid  = threadIdx.x;
    const int lane = tid & 31;
    const int wave = tid >> 5;

    float4 v0 = p[tid * 2];
    float4 v1 = p[tid * 2 + 1];

    float m = fmaxf(fmaxf(fmaxf(v0.x, v0.y), fmaxf(v0.z, v0.w)),
                    fmaxf(fmaxf(v1.x, v1.y), fmaxf(v1.z, v1.w)));
    #pragma unroll
    for (int off = 16; off > 0; off >>= 1)
        m = fmaxf(m, __shfl_xor(m, off, 32));
    if (lane == 0) red[wave] = m;
    __syncthreads();
    m = red[0];
    #pragma unroll
    for (int w = 1; w < 8; ++w) m = fmaxf(m, red[w]);
    __syncthreads();

    v0.x = __expf(v0.x - m); v0.y = __expf(v0.y - m);
    v0.z = __expf(v0.z - m); v0.w = __expf(v0.w - m);
    v1.x = __expf(v1.x - m); v1.y = __expf(v1.y - m);
    v1.z = __expf(v1.z - m); v1.w = __expf(v1.w - m);

    float s = (v0.x + v0.y + v0.z + v0.w) + (v1.x + v1.y + v1.z + v1.w);
    #pragma unroll
    for (int off = 16; off > 0; off >>= 1)
        s += __shfl_xor(s, off, 32);
    if (lane == 0) red[wave] = s;
    __syncthreads();
    s = red[0];
    #pragma unroll
    for (int w = 1; w < 8; ++w) s += red[w];

    const float inv = 1.0f / s;
    v0.x *= inv; v0.y *= inv; v0.z *= inv; v0.w *= inv;
    v1.x *= inv; v1.y *= inv; v1.z *= inv; v1.w *= inv;
    p[tid * 2]     = v0;
    p[tid * 2 + 1] = v1;
}

// ---------------------------------------------------------------------------
extern "C" void kernel_launch(void* const* d_in, const int* in_sizes, int n_in,
                              void* d_out, int out_size, void* d_ws, size_t ws_size,
                              hipStream_t stream)
{
    (void)in_sizes; (void)n_in; (void)out_size; (void)ws_size;

    const float* x  = (const float*)d_in[0];
    const float* wq = (const float*)d_in[1];
    const float* bq = (const float*)d_in[2];
    const float* wk = (const float*)d_in[3];
    const float* bk = (const float*)d_in[4];
    const float* wv = (const float*)d_in[5];
    const float* bv = (const float*)d_in[6];
    const float* wo = (const float*)d_in[7];
    const float* bo = (const float*)d_in[8];

    float* out  = (float*)d_out;                        // [B,S,D] = 4096x1024
    float* attn = out + (long)ROWS * DM;                // [B,H,S,S]

    const long NQ = (long)ROWS * DM;                    // 4M floats = 16 MB
    float* Q   = (float*)d_ws;
    float* Kb  = Q + NQ;
    float* V   = Kb + NQ;
    float* ctx = Q;                                     // Q dead after logits -> reuse

    const long SD  = (long)S_LEN * DM;                  // per-batch stride in Q/K/V
    const long SS  = (long)S_LEN * S_LEN;               // per-head stride in attn
    const dim3 blk(128);

    // 1) Q/K/V projections: [4096x1024] = x[4096x1024] @ W[1024x1024] + b
    const dim3 gproj(DM / 64, ROWS / 64, 1);
    gemm_wmma_f32<false><<<gproj, blk, 0, stream>>>(x, DM, 0, 0, wq, DM, 0, 0,
                                                    Q,  DM, 0, 0, bq, 1.0f,
                                                    ROWS, DM, DM, 1);
    gemm_wmma_f32<false><<<gproj, blk, 0, stream>>>(x, DM, 0, 0, wk, DM, 0, 0,
                                                    Kb, DM, 0, 0, bk, 1.0f,
                                                    ROWS, DM, DM, 1);
    gemm_wmma_f32<false><<<gproj, blk, 0, stream>>>(x, DM, 0, 0, wv, DM, 0, 0,
                                                    V,  DM, 0, 0, bv, 1.0f,
                                                    ROWS, DM, DM, 1);

    // 2) logits[b,h] = Q_h @ K_h^T * rsqrt(64)   (NT GEMM, 2048x2048, K=64)
    const dim3 glog(S_LEN / 64, S_LEN / 64, BATCH * NHEADS);
    gemm_wmma_f32<true><<<glog, blk, 0, stream>>>(Q,  DM, SD, 64,
                                                  Kb, DM, SD, 64,
                                                  attn, S_LEN, (long)NHEADS * SS, SS,
                                                  nullptr, 0.125f,
                                                  S_LEN, S_LEN, 64, NHEADS);

    // 3) softmax rows (in place in attn output region)
    softmax_rows<<<BATCH * NHEADS * S_LEN, 256, 0, stream>>>(attn);

    // 4) ctx[b,h] = attn @ V_h  (NN GEMM, 2048x64, K=2048), concat layout
    const dim3 gctx(1, S_LEN / 64, BATCH * NHEADS);
    gemm_wmma_f32<false><<<gctx, blk, 0, stream>>>(attn, S_LEN, (long)NHEADS * SS, SS,
                                                   V,   DM, SD, 64,
                                                   ctx, DM, SD, 64,
                                                   nullptr, 1.0f,
                                                   S_LEN, 64, S_LEN, NHEADS);

    // 5) out = ctx @ wo + bo
    gemm_wmma_f32<false><<<gproj, blk, 0, stream>>>(ctx, DM, 0, 0, wo, DM, 0, 0,
                                                    out, DM, 0, 0, bo, 1.0f,
                                                    ROWS, DM, DM, 1);
}